// RWEqualDiceLoss_50534585204960
// MI455X (gfx1250) — compile-verified
//
#include <hip/hip_runtime.h>
#include <math.h>

// Problem constants from the reference
#define Bn   8
#define Cn   4
#define Hn   256
#define Wn   256
#define HWn  (Hn * Wn)
#define BIG  512            // h + w, matches reference "big"
#define NSLICE (Bn * Cn)    // 32

typedef __attribute__((ext_vector_type(2))) float v2f;
typedef __attribute__((ext_vector_type(8))) float v8f;

// Exact brute-force min-plus for one row: min_jp ( G2[i][jp] + (j-jp)^2 )
// G2 holds pre-squared column distances -> 1 v_mad_u32 + 1 v_min per element.
__device__ __forceinline__ int minplus_row(const unsigned int* __restrict__ Gi2, int j) {
  int best = 0x7fffffff;
#pragma unroll 16
  for (int jp = 0; jp < Wn; ++jp) {
    int dj = j - jp;
    int v  = dj * dj + (int)Gi2[jp];   // broadcast LDS read (all lanes same address)
    best = min(best, v);
  }
  return best;
}

__global__ __launch_bounds__(256)
void rw_slice_kernel(const float* __restrict__ x,
                     const int*   __restrict__ yidx,
                     float*       __restrict__ part) {
  // Full per-slice squared-distance field in LDS: 256 KB of the 320 KB WGP pool.
  __shared__ unsigned int G2[Hn * Wn];
  __shared__ float wred[8];
  __shared__ float wnum[8], wden[8], wrw[8];

  const int j    = threadIdx.x;        // this thread owns column j
  const int s    = blockIdx.x;         // slice = (b,c)
  const int b    = s / Cn;
  const int c    = s % Cn;
  const int lane = j & 31;
  const int wave = j >> 5;

  const int* __restrict__ yb = yidx + (size_t)b * HWn;

  // ---------------- Phase 1: per-column 1D distance to nearest zero ----------------
  // forward scan (top -> bottom). d==0 iff y==0 at the pixel.
  int d = BIG;
  for (int i = 0; i < Hn; ++i) {
    int one = (yb[i * Wn + j] == c) ? 1 : 0;
    d = one ? min(d + 1, BIG) : 0;
    G2[i * Wn + j] = (unsigned int)d;
  }
  // backward scan; forward G==0 <=> zero pixel, so 'one' is recoverable from G.
  // Square in the same pass (column is private to this thread, no barrier yet).
  d = BIG;
  for (int i = Hn - 1; i >= 0; --i) {
    int cur = (int)G2[i * Wn + j];
    int one = (cur != 0) ? 1 : 0;
    d = one ? min(d + 1, BIG) : 0;
    int g = min(cur, d);
    G2[i * Wn + j] = (unsigned int)(g * g);
  }
  __syncthreads();

  // ---------------- Pass A: max over slice of D^2 (no D storage needed) ----------------
  float maxD2 = 0.0f;
  for (int i = 0; i < Hn; ++i) {
    int best = minplus_row(&G2[i * Wn], j);
    maxD2 = fmaxf(maxD2, (float)best);
  }
#pragma unroll
  for (int off = 16; off > 0; off >>= 1)
    maxD2 = fmaxf(maxD2, __shfl_xor(maxD2, off, 32));
  if (lane == 0) wred[wave] = maxD2;
  __syncthreads();
  float m = wred[0];
#pragma unroll
  for (int w = 1; w < 8; ++w) m = fmaxf(m, wred[w]);
  const float dmax = sqrtf(m);
  const float inv  = 1.0f / (dmax + 1e-15f);

  // ---------------- Pass B: recompute D^2, fuse r / softmax / dice / rw ----------------
  const float* __restrict__ xb = x + (size_t)b * Cn * HWn;
  float numL = 0.0f, denL = 0.0f;

  // rw accumulation through WMMA: D = A(16x4) * ones(4x16) + C, f32 exact.
  v8f acc = {};
  v2f bones; bones[0] = 1.0f; bones[1] = 1.0f;

  for (int i = 0; i < Hn; ++i) {
    const unsigned int* __restrict__ Gi2 = &G2[i * Wn];
    int best = minplus_row(Gi2, j);
    float r = (best == 0) ? 1.0f : -(sqrtf((float)best) * inv);

    const int idx = i * Wn + j;
    float x0 = xb[idx];
    float x1 = xb[HWn + idx];
    float x2 = xb[2 * HWn + idx];
    float x3 = xb[3 * HWn + idx];
    if (i + 1 < Hn) __builtin_prefetch(xb + idx + Wn, 0, 0);  // global_prefetch_b8

    float mx = fmaxf(fmaxf(x0, x1), fmaxf(x2, x3));
    float e0 = expf(x0 - mx);
    float e1 = expf(x1 - mx);
    float e2 = expf(x2 - mx);
    float e3 = expf(x3 - mx);
    float esum = e0 + e1 + e2 + e3;
    float ec   = (c == 0) ? e0 : ((c == 1) ? e1 : ((c == 2) ? e2 : e3));
    float xsv  = ec / esum;

    float onef = (Gi2[j] != 0u) ? 1.0f : 0.0f;   // one-hot mask (squaring keeps zeroness)
    numL += xsv * onef;
    denL += xsv + onef;

    v2f a; a[0] = xsv * r; a[1] = 0.0f;
    // 8-arg form: (neg_a, A, neg_b, B, c_mod, C, reuse_a, reuse_b)
    acc = __builtin_amdgcn_wmma_f32_16x16x4_f32(
        false, a, false, bones, (short)0, acc, false, false);
  }

  // dice partial reduction per wave
#pragma unroll
  for (int off = 16; off > 0; off >>= 1) {
    numL += __shfl_down(numL, off, 32);
    denL += __shfl_down(denL, off, 32);
  }
  // WMMA accumulator extraction: lanes 0-15 hold rows 0..7, lanes 16-31 rows 8..15
  float half = acc[0] + acc[1] + acc[2] + acc[3] + acc[4] + acc[5] + acc[6] + acc[7];
  float wtot = half + __shfl_xor(half, 16, 32);

  if (lane == 0) { wnum[wave] = numL; wden[wave] = denL; wrw[wave] = wtot; }
  __syncthreads();
  if (j == 0) {
    float n = 0.0f, dn = 0.0f, rwv = 0.0f;
#pragma unroll
    for (int w = 0; w < 8; ++w) { n += wnum[w]; dn += wden[w]; rwv += wrw[w]; }
    part[s * 3 + 0] = 2.0f * n;   // dice numerator
    part[s * 3 + 1] = dn;         // dice denominator
    part[s * 3 + 2] = rwv;        // sum(xs * r) over slice
  }
}

__global__ void rw_finalize_kernel(const float* __restrict__ part,
                                   float* __restrict__ out) {
  if (threadIdx.x == 0 && blockIdx.x == 0) {
    float dice = 0.0f, rw = 0.0f;
    for (int s = 0; s < NSLICE; ++s) {
      dice += part[s * 3 + 0] / (part[s * 3 + 1] + 1e-6f);
      rw   += part[s * 3 + 2];
    }
    out[0] = (1.0f - dice / (float)NSLICE) + rw / (float)(NSLICE * HWn);
  }
}

extern "C" void kernel_launch(void* const* d_in, const int* in_sizes, int n_in,
                              void* d_out, int out_size, void* d_ws, size_t ws_size,
                              hipStream_t stream) {
  const float* x = (const float*)d_in[0];   // [8,4,256,256] f32
  const int*   y = (const int*)d_in[1];     // [8,1,256,256] int labels 0..3
  (void)in_sizes; (void)n_in; (void)out_size; (void)ws_size;

  float* part = (float*)d_ws;               // 32 slices x {num, den, rwsum}
  float* out  = (float*)d_out;              // single f32 scalar

  hipLaunchKernelGGL(rw_slice_kernel, dim3(NSLICE), dim3(256), 0, stream, x, y, part);
  hipLaunchKernelGGL(rw_finalize_kernel, dim3(1), dim3(32), 0, stream, part, out);
}